// GatedGraphConvolution_63015760167426
// MI455X (gfx1250) — compile-verified
//
#include <hip/hip_runtime.h>
#include <math.h>

// ---------------------------------------------------------------------------
// GatedGraphConvolution for MI455X (gfx1250, wave32, WMMA, async-LDS DMA)
//   support = x @ w1
//   trans   = x @ w2 + b2            (written into d_out)
//   gate    = sigmoid(x @ w3 + b3)
//   agg     = segment_sum(support[edge_col] * edge_val, edge_row)
//   out     = trans + gate * (relu(agg + b1) - trans)
// ---------------------------------------------------------------------------

#define D_DIM 512
#define KC    32
#define KP    (KC + 4)     // padded K stride: 36 floats = 144 B
                           //  -> 16B-aligned rows for async b128 LDS stores
                           //  -> 8B-aligned (row, even-k) for ds_load_b64 pairs
#define BM    64
#define BN    64

typedef __attribute__((ext_vector_type(2))) float v2f;
typedef __attribute__((ext_vector_type(8))) float v8f;

// ---------------------------------------------------------------------------
// Kernel 1: fused triple GEMM with V_WMMA_F32_16X16X4_F32.
// x tile staged by the CDNA5 async Global->LDS DMA (ASYNCcnt); weight tiles
// staged manually because they are transposed into LDS (DMA cannot transpose).
// grid = (D/BN, ceil(n/BM)), block = 256 threads (8 waves)
// ---------------------------------------------------------------------------
__global__ __launch_bounds__(256, 2)
void ggc_gemm3_wmma(const float* __restrict__ x,
                    const float* __restrict__ w1,
                    const float* __restrict__ w2,
                    const float* __restrict__ w3,
                    const float* __restrict__ b2,
                    const float* __restrict__ b3,
                    float* __restrict__ support,
                    float* __restrict__ trans,
                    float* __restrict__ gate,
                    int n_rows)
{
    __shared__ __align__(16) float xs[BM][KP];         // 64 x 36, row-major (m, k)
    __shared__ __align__(16) float wtT[3][BN][KP];     // 3 x 64 x 36, TRANSPOSED (n, k)

    const int tid   = threadIdx.x;
    const int lane  = tid & 31;
    const int wave  = tid >> 5;
    const int wm    = wave >> 1;              // 0..3  (M sub-tile)
    const int wn    = wave & 1;               // 0..1  (N sub-tile)
    const int lhalf = lane >> 4;              // 0 / 1 (lane half)
    const int l15   = lane & 15;

    const int block_m = blockIdx.y * BM;
    const int block_n = blockIdx.x * BN;

    const float* wptr[3] = { w1, w2, w3 };

    v8f acc[2][3];
    {
        v8f z = {};
#pragma unroll
        for (int s = 0; s < 2; ++s)
#pragma unroll
            for (int g = 0; g < 3; ++g)
                acc[s][g] = z;
    }

    // Per-thread async-DMA slot for the x tile: 512 x (16B) slots, 2/thread.
    // slot -> (row r = slot/8, 16B chunk c4 = slot%8) of the 64x32 x tile.
    // OOB rows are clamped (their WMMA results are never stored).
    const int slot0 = tid;          // i = 0
    const int slot1 = 256 + tid;    // i = 1
    const int r0 = slot0 >> 3, c40 = slot0 & 7;
    const int r1 = slot1 >> 3, c41 = slot1 & 7;
    int grow0 = block_m + r0; if (grow0 > n_rows - 1) grow0 = n_rows - 1;
    int grow1 = block_m + r1; if (grow1 > n_rows - 1) grow1 = n_rows - 1;
    // AS(3) offsets: low 32 bits of the flat address of the LDS destination.
    const unsigned lds0 = (unsigned)(size_t)&xs[r0][c40 * 4];
    const unsigned lds1 = (unsigned)(size_t)&xs[r1][c41 * 4];

    for (int k0 = 0; k0 < D_DIM; k0 += KC) {
        // ---- x tile: async Global->LDS DMA (b128/lane), no VGPR round-trip
        {
            const float* g0 = &x[(size_t)grow0 * D_DIM + k0 + c40 * 4];
            const float* g1 = &x[(size_t)grow1 * D_DIM + k0 + c41 * 4];
            asm volatile("global_load_async_to_lds_b128 %0, %1, off"
                         :: "v"(lds0), "v"(g0) : "memory");
            asm volatile("global_load_async_to_lds_b128 %0, %1, off"
                         :: "v"(lds1), "v"(g1) : "memory");
        }
        // ---- stage 3 weight tiles transposed: (k,n) global -> (n,k) LDS ----
        // (overlaps with the async x DMA above)
#pragma unroll
        for (int g = 0; g < 3; ++g) {
            const float* wsrc = wptr[g];
#pragma unroll
            for (int i = 0; i < 2; ++i) {
                int slot = i * 256 + tid;      // 0..511
                int kr   = slot >> 4;          // 0..31
                int c4   = slot & 15;          // 0..15
                float4 v = *reinterpret_cast<const float4*>(
                    &wsrc[(size_t)(k0 + kr) * D_DIM + block_n + c4 * 4]);
                wtT[g][c4 * 4 + 0][kr] = v.x;  // transpose scatter
                wtT[g][c4 * 4 + 1][kr] = v.y;
                wtT[g][c4 * 4 + 2][kr] = v.z;
                wtT[g][c4 * 4 + 3][kr] = v.w;
            }
        }
        // Drain this wave's async DMA before the workgroup barrier.
        asm volatile("s_wait_asynccnt 0x0" ::: "memory");
        __syncthreads();

        // ---- 8 x (K=4) WMMA steps over this K-chunk ------------------------
#pragma unroll
        for (int kk = 0; kk < KC; kk += 4) {
            const int ka = kk + lhalf * 2;     // even -> 8B-aligned pair loads
            // A: 16x4 f32, lane = M, pair = (K, K+1), lane-half selects K+2
            const v2f a = *reinterpret_cast<const v2f*>(&xs[wm * 16 + l15][ka]);
#pragma unroll
            for (int s = 0; s < 2; ++s) {
                const int ncol = wn * 32 + s * 16 + l15;
#pragma unroll
                for (int g = 0; g < 3; ++g) {
                    // B: 4x16 f32, lane = N, pair contiguous in transposed LDS
                    const v2f b = *reinterpret_cast<const v2f*>(&wtT[g][ncol][ka]);
                    acc[s][g] = __builtin_amdgcn_wmma_f32_16x16x4_f32(
                        /*neg_a=*/false, a, /*neg_b=*/false, b,
                        /*c_mod=*/(short)0, acc[s][g],
                        /*reuse_a=*/false, /*reuse_b=*/false);
                }
            }
        }
        __syncthreads();
    }

    // ---- store: C/D layout — VGPR r holds row r + 8*lanehalf, col = lane&15
#pragma unroll
    for (int s = 0; s < 2; ++s) {
        const int col = block_n + wn * 32 + s * 16 + l15;
        const float bias2 = b2[col];
        const float bias3 = b3[col];
#pragma unroll
        for (int r = 0; r < 8; ++r) {
            const int row = block_m + wm * 16 + lhalf * 8 + r;
            if (row < n_rows) {
                const size_t idx = (size_t)row * D_DIM + col;
                support[idx] = acc[s][0][r];
                trans[idx]   = acc[s][1][r] + bias2;
                const float gz = acc[s][2][r] + bias3;
                gate[idx]    = 1.0f / (1.0f + __expf(-gz));
            }
        }
    }
}

// ---------------------------------------------------------------------------
// Kernel 2: zero agg buffer (float4 grid-stride)
// ---------------------------------------------------------------------------
__global__ void ggc_zero(float4* __restrict__ p, size_t n4)
{
    size_t i      = (size_t)blockIdx.x * blockDim.x + threadIdx.x;
    size_t stride = (size_t)gridDim.x * blockDim.x;
    for (; i < n4; i += stride)
        p[i] = make_float4(0.f, 0.f, 0.f, 0.f);
}

// ---------------------------------------------------------------------------
// Kernel 3: edge scatter — one wave32 per edge.
// Gather support[col] (float4), scale by edge_val, atomicAdd into agg[row].
// Prefetch the next edge's source row (global_prefetch_b8).
// ---------------------------------------------------------------------------
__global__ __launch_bounds__(256)
void ggc_scatter(const float* __restrict__ support,
                 const int*   __restrict__ erow,
                 const int*   __restrict__ ecol,
                 const float* __restrict__ eval,
                 float*       __restrict__ agg,
                 int nedges)
{
    const int lane   = threadIdx.x & 31;
    const int wid    = (int)((blockIdx.x * blockDim.x + threadIdx.x) >> 5);
    const int nwaves = (int)((gridDim.x * blockDim.x) >> 5);

    for (int e = wid; e < nedges; e += nwaves) {
        const int   row = erow[e];
        const int   col = ecol[e];
        const float val = eval[e];

        // Prefetch next edge's gather row: 32 lanes x 64B covers the 2KB row.
        if (e + nwaves < nedges) {
            const int ncol = ecol[e + nwaves];
            __builtin_prefetch(&support[(size_t)ncol * D_DIM + lane * 16], 0, 1);
        }

        const float* __restrict__ src = support + (size_t)col * D_DIM;
        float*       __restrict__ dst = agg     + (size_t)row * D_DIM;

#pragma unroll
        for (int d = lane * 4; d < D_DIM; d += 128) {
            const float4 v = *reinterpret_cast<const float4*>(src + d);
            atomicAdd(&dst[d + 0], v.x * val);
            atomicAdd(&dst[d + 1], v.y * val);
            atomicAdd(&dst[d + 2], v.z * val);
            atomicAdd(&dst[d + 3], v.w * val);
        }
    }
}

// ---------------------------------------------------------------------------
// Kernel 4: epilogue — out = trans + gate * (relu(agg + b1) - trans), in place
// ---------------------------------------------------------------------------
__global__ void ggc_finalize(const float* __restrict__ agg,
                             const float* __restrict__ b1,
                             const float* __restrict__ gate,
                             float*       __restrict__ out,
                             size_t total)
{
    size_t i      = (size_t)blockIdx.x * blockDim.x + threadIdx.x;
    size_t stride = (size_t)gridDim.x * blockDim.x;
    for (; i < total; i += stride) {
        const int   col = (int)(i & (D_DIM - 1));
        float o = agg[i] + b1[col];
        o = fmaxf(o, 0.0f);
        const float t = out[i];
        out[i] = t + gate[i] * (o - t);
    }
}

// ---------------------------------------------------------------------------
extern "C" void kernel_launch(void* const* d_in, const int* in_sizes, int n_in,
                              void* d_out, int out_size, void* d_ws, size_t ws_size,
                              hipStream_t stream)
{
    const float* x    = (const float*)d_in[0];
    const float* w1   = (const float*)d_in[1];
    const float* w2   = (const float*)d_in[2];
    const float* w3   = (const float*)d_in[3];
    const float* b1   = (const float*)d_in[4];
    const float* b2   = (const float*)d_in[5];
    const float* b3   = (const float*)d_in[6];
    const int*   erow = (const int*)d_in[7];
    const int*   ecol = (const int*)d_in[8];
    const float* eval = (const float*)d_in[9];

    const int    n      = in_sizes[0] / D_DIM;
    const int    nedges = in_sizes[7];
    const size_t total  = (size_t)n * D_DIM;

    float* support = (float*)d_ws;             // N*512 f32
    float* agg     = support + total;          // N*512 f32
    float* gate    = agg + total;              // N*512 f32
    float* out     = (float*)d_out;            // trans lives here, then result

    // 1) fused triple GEMM (WMMA f32 + async-LDS staging)
    {
        dim3 grid(D_DIM / BN, (n + BM - 1) / BM);
        ggc_gemm3_wmma<<<grid, 256, 0, stream>>>(x, w1, w2, w3, b2, b3,
                                                 support, out, gate, n);
    }
    // 2) zero agg
    {
        const size_t n4 = total / 4;
        int blocks = (int)((n4 + 255) / 256);
        if (blocks > 4096) blocks = 4096;
        ggc_zero<<<blocks, 256, 0, stream>>>((float4*)agg, n4);
    }
    // 3) edge gather + atomic scatter
    {
        ggc_scatter<<<4096, 256, 0, stream>>>(support, erow, ecol, eval,
                                              agg, nedges);
    }
    // 4) epilogue
    {
        int blocks = (int)((total + 255) / 256);
        if (blocks > 8192) blocks = 8192;
        ggc_finalize<<<blocks, 256, 0, stream>>>(agg, b1, gate, out, total);
    }
}